// NodeAttnModel_11209864642801
// MI455X (gfx1250) — compile-verified
//
#include <hip/hip_runtime.h>
#include <math.h>

#define Nn   50000
#define Dd   160
#define Hh   5
#define Cc   32
#define HC   160
#define Ee   800000
#define EDd  16
#define HIDn 512
#define EPSf 1e-5f
#define NEGs 0.2f

typedef float v2f __attribute__((ext_vector_type(2)));
typedef float v8f __attribute__((ext_vector_type(8)));
typedef int   v4i __attribute__((ext_vector_type(4)));
typedef int   v8i __attribute__((ext_vector_type(8)));

// ---- fp32 WMMA 16x16x4 wrapper (D = A(16x4) * B(4x16) + C) ----
__device__ __forceinline__ v8f wmma4(v2f a, v2f b, v8f c) {
#if defined(__HIP_DEVICE_COMPILE__)
# if __has_builtin(__builtin_amdgcn_wmma_f32_16x16x4_f32)
  return __builtin_amdgcn_wmma_f32_16x16x4_f32(false, a, false, b, (short)0, c,
                                               false, false);
# else
#  error "__builtin_amdgcn_wmma_f32_16x16x4_f32 not available on this toolchain"
# endif
#else
  (void)a; (void)b;
  return c;
#endif
}

// ---- Tensor Data Mover: contiguous n_f32-float block, global -> LDS ----
// D# per CDNA5 ISA: count=1, type=2 (image),
// data_size=4B, 1-row tile: tile_dim0=n, tile_dim1=1, tensor_dim0=n,
// tensor_dim1=1, stride0=n. Issue from ONE wave only; completion via TENSORcnt.
__device__ __forceinline__ void tdm_load_contig(void* ldsPtr, const void* gPtr,
                                                int n_f32) {
#if defined(__HIP_DEVICE_COMPILE__)
  unsigned lds = (unsigned)(unsigned long long)ldsPtr;   // flat low32 == LDS offset
  unsigned long long ga = (unsigned long long)gPtr;
  v4i g0;
  g0.x = 1;                                              // count=1 valid descriptor
  g0.y = (int)lds;                                       // lds_addr
  g0.z = (int)(unsigned)(ga & 0xffffffffull);            // global_addr[31:0]
  g0.w = (int)((unsigned)((ga >> 32) & 0x1ffffffull) | (2u << 30)); // [56:32]|type=2
  v8i g1;
  g1[0] = 0x00020000;                         // wg_mask=0, data_size=2 (4 bytes)
  g1[1] = (n_f32 & 0xffff) << 16;             // tensor_dim0[15:0] @ bit48
  g1[2] = ((n_f32 >> 16) & 0xffff) | (1 << 16); // tensor_dim0[31:16], tensor_dim1=1
  g1[3] = (n_f32 & 0xffff) << 16;             // tile_dim0 @ bit112 (16-bit)
  g1[4] = 1;                                  // tile_dim1=1, tile_dim2=0
  g1[5] = n_f32;                              // tensor_dim0_stride[31:0]
  g1[6] = 0;                                  // stride0 hi, stride1 lo
  g1[7] = 0;                                  // stride1 hi
  asm volatile("tensor_load_to_lds %0, %1" :: "s"(g0), "s"(g1) : "memory");
#else
  (void)ldsPtr; (void)gPtr; (void)n_f32;
#endif
}

__device__ __forceinline__ void tdm_wait0() {
#if defined(__HIP_DEVICE_COMPILE__)
  __builtin_amdgcn_s_wait_tensorcnt(0);
#endif
}

__device__ __forceinline__ float wred(float v) {
#pragma unroll
  for (int o = 16; o > 0; o >>= 1) v += __shfl_xor(v, o, 32);
  return v;
}

__device__ __forceinline__ void atomicMaxF(float* a, float v) {
  int old = __float_as_int(*a);
  while (__int_as_float(old) < v) {
    int prev = atomicCAS((int*)a, old, __float_as_int(v));
    if (prev == old) break;
    old = prev;
  }
}

// ---- init: mx = -inf, denom = 0 ----
__global__ void k_init(float* __restrict__ mx, float* __restrict__ denom) {
  int i = blockIdx.x * blockDim.x + threadIdx.x;
  if (i < Nn * Hh) { mx[i] = -INFINITY; denom[i] = 0.f; }
}

__global__ void k_zero(float* __restrict__ p, int n) {
  int i = blockIdx.x * blockDim.x + threadIdx.x;
  if (i < n) p[i] = 0.f;
}

// ---- node projections: xl = x@Wl+bl, xr = x@Wr+br (WMMA, 16-row tiles) ----
__global__ __launch_bounds__(256) void k_proj(
    const float* __restrict__ x,
    const float* __restrict__ Wl, const float* __restrict__ bl,
    const float* __restrict__ Wr, const float* __restrict__ br,
    float* __restrict__ xl, float* __restrict__ xr) {
  __shared__ float sx[16 * Dd];
  const int n0 = blockIdx.x * 16;
  if (threadIdx.x < 32) {                       // wave 0 drives the TDM
    tdm_load_contig(sx, x + n0 * Dd, 16 * Dd);
    tdm_wait0();
  }
  __syncthreads();
  const int lane = threadIdx.x & 31, w = threadIdx.x >> 5;
  const int m = lane & 15, kh = (lane >> 4) << 1, mh = (lane >> 4) << 3;
  for (int t = w; t < 20; t += 8) {             // 10 n-tiles x 2 matrices
    const int mat = t / 10, nt = t % 10, n = nt * 16 + m;
    const float* W   = mat ? Wr : Wl;
    const float* bia = mat ? br : bl;
    float*       dst = mat ? xr : xl;
    v8f acc = {};
    for (int k = 0; k < Dd; k += 4) {
      v2f a, b;
      a.x = sx[m * Dd + k + kh];       a.y = sx[m * Dd + k + kh + 1];
      b.x = W[(k + kh) * HC + n];      b.y = W[(k + kh + 1) * HC + n];
      acc = wmma4(a, b, acc);
    }
    const float bn = bia[n];
#pragma unroll
    for (int r = 0; r < 8; r++) dst[(n0 + mh + r) * HC + n] = acc[r] + bn;
  }
}

// ---- edge scores: score[e,h] = att_h . lrelu(xl[src]+xr[dst]+ea), atomic max ----
__global__ __launch_bounds__(256) void k_score(
    const float* __restrict__ xl, const float* __restrict__ xr,
    const float* __restrict__ eattr, const int* __restrict__ ei,
    const float* __restrict__ We, const float* __restrict__ att,
    float* __restrict__ score, float* __restrict__ mx) {
  __shared__ float sW[EDd * HC];
  __shared__ float sA[HC];
  if (threadIdx.x < 32) {                       // stage We + att via TDM
    tdm_load_contig(sW, We, EDd * HC);
    tdm_load_contig(sA, att, HC);
    tdm_wait0();
  }
  __syncthreads();
  const int lane = threadIdx.x & 31;
  const int wid  = (blockIdx.x * 256 + threadIdx.x) >> 5;
  const int nw   = (gridDim.x * 256) >> 5;
  for (int e = wid; e < Ee; e += nw) {
    const int src = ei[e], dst = ei[Ee + e];
    float ek[EDd];
    const float4* ep = (const float4*)(eattr + e * EDd);
    float4 q;
    q = ep[0]; ek[0]=q.x; ek[1]=q.y; ek[2]=q.z; ek[3]=q.w;
    q = ep[1]; ek[4]=q.x; ek[5]=q.y; ek[6]=q.z; ek[7]=q.w;
    q = ep[2]; ek[8]=q.x; ek[9]=q.y; ek[10]=q.z; ek[11]=q.w;
    q = ep[3]; ek[12]=q.x; ek[13]=q.y; ek[14]=q.z; ek[15]=q.w;
#pragma unroll
    for (int j = 0; j < Hh; j++) {
      const int c = j * 32 + lane;              // head j == lane-block j
      float ea = 0.f;
#pragma unroll
      for (int k = 0; k < EDd; k++) ea = fmaf(ek[k], sW[k * HC + c], ea);
      float v = xl[src * HC + c] + xr[dst * HC + c] + ea;
      v = v > 0.f ? v : NEGs * v;
      float s = wred(v * sA[c]);
      if (lane == 0) {
        score[e * Hh + j] = s;
        atomicMaxF(&mx[dst * Hh + j], s);
      }
    }
  }
}

// ---- exp + denom ----
__global__ void k_exp(float* __restrict__ score, const int* __restrict__ ei,
                      const float* __restrict__ mx, float* __restrict__ denom) {
  int i = blockIdx.x * blockDim.x + threadIdx.x;
  if (i < Ee * Hh) {
    int e = i / Hh, h = i - e * Hh;
    int dst = ei[Ee + e];
    float ex = __expf(score[i] - mx[dst * Hh + h]);
    score[i] = ex;
    atomicAdd(&denom[dst * Hh + h], ex);
  }
}

// ---- aggregate: agg[dst] += alpha * xl[src] ----
__global__ __launch_bounds__(256) void k_agg(
    const float* __restrict__ xl, const float* __restrict__ score,
    const float* __restrict__ denom, const int* __restrict__ ei,
    float* __restrict__ agg) {
  const int lane = threadIdx.x & 31;
  const int wid  = (blockIdx.x * 256 + threadIdx.x) >> 5;
  const int nw   = (gridDim.x * 256) >> 5;
  for (int e = wid; e < Ee; e += nw) {
    const int src = ei[e], dst = ei[Ee + e];
#pragma unroll
    for (int j = 0; j < Hh; j++) {
      float al = score[e * Hh + j] / denom[dst * Hh + j];
      int c = j * 32 + lane;
      atomicAdd(&agg[dst * HC + c], al * xl[src * HC + c]);
    }
  }
}

// ---- out = LN(x + agg + b_gat) ----
__global__ __launch_bounds__(256) void k_ln1(
    const float* __restrict__ x, const float* __restrict__ agg,
    const float* __restrict__ bgat, const float* __restrict__ g1,
    const float* __restrict__ b1, float* __restrict__ out) {
  const int lane = threadIdx.x & 31;
  const int wid  = (blockIdx.x * 256 + threadIdx.x) >> 5;
  const int nw   = (gridDim.x * 256) >> 5;
  for (int n = wid; n < Nn; n += nw) {
    float v[5], s = 0.f, s2 = 0.f;
#pragma unroll
    for (int j = 0; j < 5; j++) {
      int c = j * 32 + lane;
      float t = x[n * Dd + c] + agg[n * Dd + c] + bgat[c];
      v[j] = t; s += t; s2 += t * t;
    }
    s = wred(s); s2 = wred(s2);
    float mean = s * (1.f / 160.f), var = s2 * (1.f / 160.f) - mean * mean;
    float rs = rsqrtf(var + EPSf);
#pragma unroll
    for (int j = 0; j < 5; j++) {
      int c = j * 32 + lane;
      out[n * Dd + c] = (v[j] - mean) * rs * g1[c] + b1[c];
    }
  }
}

// ---- fused MLP: LN(selu(out@W1+b1)) @ W2 + b2, residual, final LN (WMMA) ----
__global__ __launch_bounds__(256) void k_mlp(
    const float* __restrict__ outb,
    const float* __restrict__ W1, const float* __restrict__ b1m,
    const float* __restrict__ gm, const float* __restrict__ bm,
    const float* __restrict__ W2, const float* __restrict__ b2m,
    const float* __restrict__ g2, const float* __restrict__ b2v,
    float* __restrict__ y) {
  __shared__ float so[16 * Dd];     // out tile
  __shared__ float sh[16 * HIDn];   // hidden tile
  __shared__ float sm[16 * Dd];     // mlp tile
  const int n0 = blockIdx.x * 16;
  if (threadIdx.x < 32) {                       // TDM-stage the out tile
    tdm_load_contig(so, outb + n0 * Dd, 16 * Dd);
    tdm_wait0();
  }
  __syncthreads();
  const int lane = threadIdx.x & 31, w = threadIdx.x >> 5;
  const int m = lane & 15, kh = (lane >> 4) << 1, mh = (lane >> 4) << 3;

  // GEMM1: 16x512 over 32 n-tiles, 4 per wave
  v8f acc[4]; acc[0] = {}; acc[1] = {}; acc[2] = {}; acc[3] = {};
  for (int k = 0; k < Dd; k += 4) {
    v2f a; a.x = so[m * Dd + k + kh]; a.y = so[m * Dd + k + kh + 1];
#pragma unroll
    for (int i = 0; i < 4; i++) {
      int n = (w + 8 * i) * 16 + m;
      v2f b; b.x = W1[(k + kh) * HIDn + n]; b.y = W1[(k + kh + 1) * HIDn + n];
      acc[i] = wmma4(a, b, acc[i]);
    }
  }
  const float SA = 1.6732632423543772f, SS = 1.0507009873554805f;
#pragma unroll
  for (int i = 0; i < 4; i++) {
    int n = (w + 8 * i) * 16 + m;
    float bb = b1m[n];
#pragma unroll
    for (int r = 0; r < 8; r++) {
      float t = acc[i][r] + bb;
      t = t > 0.f ? SS * t : SS * SA * (__expf(t) - 1.f);
      sh[(mh + r) * HIDn + n] = t;
    }
  }
  __syncthreads();
  // row LayerNorm over 512 (2 rows per wave)
#pragma unroll
  for (int rr = 0; rr < 2; rr++) {
    int row = w * 2 + rr;
    float s = 0.f, s2 = 0.f;
#pragma unroll
    for (int j = 0; j < 16; j++) { float t = sh[row * HIDn + j * 32 + lane]; s += t; s2 += t * t; }
    s = wred(s); s2 = wred(s2);
    float mean = s * (1.f / 512.f), var = s2 * (1.f / 512.f) - mean * mean;
    float rs = rsqrtf(var + EPSf);
#pragma unroll
    for (int j = 0; j < 16; j++) {
      int c = j * 32 + lane;
      sh[row * HIDn + c] = (sh[row * HIDn + c] - mean) * rs * gm[c] + bm[c];
    }
  }
  __syncthreads();
  // GEMM2: 16x160 over 10 n-tiles (waves 0,1 take two)
  v8f a2[2]; a2[0] = {}; a2[1] = {};
  const int ntc = (w < 2) ? 2 : 1;
  for (int k = 0; k < HIDn; k += 4) {
    v2f a; a.x = sh[m * HIDn + k + kh]; a.y = sh[m * HIDn + k + kh + 1];
    for (int i = 0; i < ntc; i++) {
      int n = (w + 8 * i) * 16 + m;
      v2f b; b.x = W2[(k + kh) * Dd + n]; b.y = W2[(k + kh + 1) * Dd + n];
      a2[i] = wmma4(a, b, a2[i]);
    }
  }
  for (int i = 0; i < ntc; i++) {
    int n = (w + 8 * i) * 16 + m;
    float bb = b2m[n];
#pragma unroll
    for (int r = 0; r < 8; r++) sm[(mh + r) * Dd + n] = a2[i][r] + bb;
  }
  __syncthreads();
  // final LN(out + mlp)
#pragma unroll
  for (int rr = 0; rr < 2; rr++) {
    int row = w * 2 + rr;
    float v[5], s = 0.f, s2 = 0.f;
#pragma unroll
    for (int j = 0; j < 5; j++) {
      int c = j * 32 + lane;
      float t = so[row * Dd + c] + sm[row * Dd + c];
      v[j] = t; s += t; s2 += t * t;
    }
    s = wred(s); s2 = wred(s2);
    float mean = s * (1.f / 160.f), var = s2 * (1.f / 160.f) - mean * mean;
    float rs = rsqrtf(var + EPSf);
#pragma unroll
    for (int j = 0; j < 5; j++) {
      int c = j * 32 + lane;
      y[(n0 + row) * Dd + c] = (v[j] - mean) * rs * g2[c] + b2v[c];
    }
  }
}

extern "C" void kernel_launch(void* const* d_in, const int* in_sizes, int n_in,
                              void* d_out, int out_size, void* d_ws, size_t ws_size,
                              hipStream_t stream) {
  const float* x    = (const float*)d_in[0];
  const int*   ei   = (const int*)d_in[1];
  const float* ea   = (const float*)d_in[2];
  // d_in[3]=u, d_in[4]=batch : unused by reference
  const float* Wl   = (const float*)d_in[5];
  const float* bl   = (const float*)d_in[6];
  const float* Wr   = (const float*)d_in[7];
  const float* br   = (const float*)d_in[8];
  const float* We   = (const float*)d_in[9];
  const float* att  = (const float*)d_in[10];
  const float* bgat = (const float*)d_in[11];
  const float* g1   = (const float*)d_in[12];
  const float* b1   = (const float*)d_in[13];
  const float* Wm1  = (const float*)d_in[14];
  const float* bm1  = (const float*)d_in[15];
  const float* gm   = (const float*)d_in[16];
  const float* bm   = (const float*)d_in[17];
  const float* Wm2  = (const float*)d_in[18];
  const float* bm2  = (const float*)d_in[19];
  const float* g2   = (const float*)d_in[20];
  const float* b2   = (const float*)d_in[21];
  float* y = (float*)d_out;

  float* xl    = (float*)d_ws;                    // N*160
  float* xr    = xl    + (size_t)Nn * HC;         // N*160 (later reused as agg)
  float* outb  = xr    + (size_t)Nn * HC;         // N*160
  float* score = outb  + (size_t)Nn * Dd;         // E*5
  float* mx    = score + (size_t)Ee * Hh;         // N*5
  float* denom = mx    + (size_t)Nn * Hh;         // N*5
  float* agg   = xr;                              // overlay (xr dead after k_score)

  k_init<<<(Nn * Hh + 255) / 256, 256, 0, stream>>>(mx, denom);
  k_proj<<<Nn / 16, 256, 0, stream>>>(x, Wl, bl, Wr, br, xl, xr);
  k_score<<<2048, 256, 0, stream>>>(xl, xr, ea, ei, We, att, score, mx);
  k_zero<<<(Nn * HC + 255) / 256, 256, 0, stream>>>(agg, Nn * HC);
  k_exp<<<(Ee * Hh + 255) / 256, 256, 0, stream>>>(score, ei, mx, denom);
  k_agg<<<2048, 256, 0, stream>>>(xl, score, denom, ei, agg);
  k_ln1<<<Nn / 8, 256, 0, stream>>>(x, agg, bgat, g1, b1, outb);
  k_mlp<<<Nn / 16, 256, 0, stream>>>(outb, Wm1, bm1, gm, bm, Wm2, bm2, g2, b2, y);
}